// MySAGPooling_27590869909662
// MI455X (gfx1250) — compile-verified
//
#include <hip/hip_runtime.h>
#include <math.h>

typedef __attribute__((ext_vector_type(2))) float v2f;
typedef __attribute__((ext_vector_type(8))) float v8f;
typedef int v4i __attribute__((ext_vector_type(4)));

// pointer-to-addrspace-qualified-int4 types for the async-to-LDS builtin
typedef __attribute__((address_space(1))) v4i* gv4i_p;   // global
typedef __attribute__((address_space(3))) v4i* lv4i_p;   // LDS

#define IN_DIM 256
#define WPB 2   // waves per block in WMMA kernel

// ---------- helpers ----------
__device__ __forceinline__ unsigned fkey(float f){
  unsigned u = __float_as_uint(f);
  return (u & 0x80000000u) ? ~u : (u | 0x80000000u);   // monotone increasing key
}
__device__ __forceinline__ float funkey(unsigned k){
  unsigned u = (k & 0x80000000u) ? (k ^ 0x80000000u) : ~k;
  return __uint_as_float(u);
}

// ---------- 0: pack 16 weight columns (K,Q,V,M x 4 types) + biases ----------
__global__ void hgt_pack_w(const float* kW, const float* kb, const float* qW, const float* qb,
                           const float* vW, const float* vb, const float* mW, const float* mb,
                           float* Wall, float* bias){
  int i = blockIdx.x * blockDim.x + threadIdx.x;
  if (i < IN_DIM * 16){
    int k = i >> 4, c = i & 15;
    int proj = c >> 2, t = c & 3;
    const float* W = (proj==0)?kW:(proj==1)?qW:(proj==2)?vW:mW;
    Wall[k*16 + c] = W[t*IN_DIM + k];
  }
  if (i < 16){
    int proj = i >> 2, t = i & 3;
    const float* B = (proj==0)?kb:(proj==1)?qb:(proj==2)?vb:mb;
    bias[i] = B[t];
  }
}

// ---------- 1: init per-node accumulators ----------
__global__ void hgt_init(unsigned* amaxKey, float* den, float* aggr,
                         unsigned char* nmask, int N){
  int i = blockIdx.x * blockDim.x + threadIdx.x;
  if (i < N){ amaxKey[i] = 0u; den[i] = 0.f; aggr[i] = 0.f; nmask[i] = 0; }
}

// ---------- 2: projection GEMM via V_WMMA_F32_16X16X4_F32 ----------
// one wave computes a 16(node) x 16(proj*4+type) f32 tile, K = 256 in steps of 4.
// x-tile is staged to LDS with GLOBAL_LOAD_ASYNC_TO_LDS_B128 (ASYNCcnt path);
// the 256x16 weight panel is staged once per block.
__global__ __launch_bounds__(WPB*32) void hgt_proj_wmma(
    const float* __restrict__ x, const int* __restrict__ ntype,
    const float* __restrict__ Wall, const float* __restrict__ bias,
    float* Kn, float* Qn, float* Vn, float* Mn, int N){
  __shared__ __align__(16) float tile[WPB][16 * IN_DIM];
  __shared__ __align__(16) float WallS[IN_DIM * 16];
  int tid  = threadIdx.x;
  int wid  = tid >> 5;
  int lane = tid & 31;
  long base = ((long)blockIdx.x * WPB + wid) * 16;   // first node of this wave's tile
  const float* src = x + base * IN_DIM;

  // whole block cooperatively stages the weight panel (4096 floats, 16KB)
  for (int i = tid * 4; i < IN_DIM * 16; i += WPB * 32 * 4)
    *(float4*)&WallS[i] = *(const float4*)&Wall[i];

  // stage 16x256 x-tile into LDS
#if __has_builtin(__builtin_amdgcn_global_load_async_to_lds_b128)
  for (int i = lane * 4; i < 16 * IN_DIM; i += 32 * 4){
    long node = base + (i >> 8);
    if (node < N){
      __builtin_amdgcn_global_load_async_to_lds_b128(
          (gv4i_p)(src + i), (lv4i_p)&tile[wid][i], 0, 0);
    } else {
      *(float4*)&tile[wid][i] = make_float4(0.f, 0.f, 0.f, 0.f);
    }
  }
#if __has_builtin(__builtin_amdgcn_s_wait_asynccnt)
  __builtin_amdgcn_s_wait_asynccnt(0);
#else
  asm volatile("s_wait_asynccnt 0x0" ::: "memory");
#endif
#else
  for (int i = lane * 4; i < 16 * IN_DIM; i += 32 * 4){
    float4 v = make_float4(0.f, 0.f, 0.f, 0.f);
    long node = base + (i >> 8);
    if (node < N) v = *(const float4*)(src + i);
    *(float4*)&tile[wid][i] = v;
  }
#endif
  __syncthreads();

  v8f acc = {};
  int row   = lane & 15;            // A row (node) / B col
  int khalf = (lane >> 4) * 2;      // K sub-offset per lane half
  for (int k = 0; k < IN_DIM; k += 4){
    int kk = k + khalf;
    v2f a, b;
    a.x = tile[wid][row * IN_DIM + kk];
    a.y = tile[wid][row * IN_DIM + kk + 1];
    b.x = WallS[kk * 16 + row];
    b.y = WallS[(kk + 1) * 16 + row];
    acc = __builtin_amdgcn_wmma_f32_16x16x4_f32(
        false, a, false, b, (short)0, acc, false, false);
  }

  int col   = row;
  int rbase = (lane < 16) ? 0 : 8;
  for (int r = 0; r < 8; ++r){
    long node = base + rbase + r;
    if (node >= N) continue;
    float val = acc[r] + bias[col];
    int t = ntype[node];
    if      (col == t     ) Kn[node] = val;
    else if (col == 4 + t ) Qn[node] = val;
    else if (col == 8 + t ) Vn[node] = val;
    else if (col == 12 + t) Mn[node] = val;
  }
}

// ---------- 3: edge pass 1 — att/msg + segment max ----------
__global__ void hgt_edge1(const int* __restrict__ src, const int* __restrict__ dst,
                          const int* __restrict__ etype,
                          const float* __restrict__ Kn, const float* __restrict__ Qn,
                          const float* __restrict__ Vn,
                          const float* __restrict__ rel_att, const float* __restrict__ rel_msg,
                          const float* __restrict__ rel_pri,
                          float* att, float* msg, unsigned* amaxKey, int E){
  int e = blockIdx.x * blockDim.x + threadIdx.x;
  if (e >= E) return;
  int s = src[e], d = dst[e], r = etype[e];
  float a = Qn[d] * Kn[s] * rel_att[r] * rel_pri[r];   // DK==1 -> /sqrt(DK)==1
  att[e] = a;
  msg[e] = Vn[s] * rel_msg[r];
  atomicMax(&amaxKey[d], fkey(a));
}

// ---------- 4: edge pass 2 — exp + segment denominator ----------
__global__ void hgt_edge2(const int* __restrict__ dst, float* att,
                          const unsigned* __restrict__ amaxKey, float* den, int E){
  int e = blockIdx.x * blockDim.x + threadIdx.x;
  if (e >= E) return;
  int d = dst[e];
  float ex = __expf(att[e] - funkey(amaxKey[d]));
  att[e] = ex;
  atomicAdd(&den[d], ex);
}

// ---------- 5: edge pass 3 — weighted aggregate ----------
__global__ void hgt_edge3(const int* __restrict__ dst, const float* __restrict__ att,
                          const float* __restrict__ msg, const float* __restrict__ den,
                          float* aggr, int E){
  int e = blockIdx.x * blockDim.x + threadIdx.x;
  if (e >= E) return;
  int d = dst[e];
  float alpha = att[e] / (den[d] + 1e-16f);
  atomicAdd(&aggr[d], msg[e] * alpha);
}

// ---------- 6: node score ----------
__global__ void hgt_score(const float* __restrict__ aggr, const int* __restrict__ ntype,
                          const float* __restrict__ aW, const float* __restrict__ ab,
                          const float* __restrict__ Mn, const float* __restrict__ skip,
                          float* score, int N){
  int n = blockIdx.x * blockDim.x + threadIdx.x;
  if (n >= N) return;
  float a = aggr[n];
  float h = 0.5f * a * (1.0f + erff(a * 0.70710678118654752f));  // exact gelu
  int t = ntype[n];
  float trans = h * aW[t] + ab[t];
  float sk = 1.0f / (1.0f + __expf(-skip[t]));
  float res = trans * sk + Mn[n] * (1.0f - sk);
  score[n] = tanhf(res);
}

// ---------- 7: build 64-bit sort keys (score desc, index asc tie-break) ----------
__global__ void hgt_fill_keys(const float* __restrict__ score,
                              unsigned long long* keys, int N, int SN){
  int i = blockIdx.x * blockDim.x + threadIdx.x;
  if (i >= SN) return;
  unsigned long long k = 0ull;
  if (i < N)
    k = ((unsigned long long)fkey(score[i]) << 32) | (unsigned)(~(unsigned)i);
  keys[i] = k;
}

// ---------- 8: bitonic sort step (overall descending) ----------
__global__ void hgt_bitonic(unsigned long long* d, int j, int k){
  int i = blockIdx.x * blockDim.x + threadIdx.x;
  int ixj = i ^ j;
  if (ixj <= i) return;
  unsigned long long a = d[i], b = d[ixj];
  bool descBlock = ((i & k) == 0);
  if (descBlock ? (a < b) : (a > b)){ d[i] = b; d[ixj] = a; }
}

// ---------- 9: emit perm/vals/masks + scalar outputs ----------
__global__ void hgt_select(const unsigned long long* __restrict__ keys,
                           const float* __restrict__ score, const int* __restrict__ ntype,
                           int* perm, float* vals, unsigned char* nmask, int* idxmap,
                           float* out_nt, float* out_batch, int kkeep){
  int i = blockIdx.x * blockDim.x + threadIdx.x;
  if (i >= kkeep) return;
  int node = (int)(~(unsigned)(keys[i] & 0xFFFFFFFFull));
  perm[i] = node;
  float v = score[node];
  vals[i] = v;
  nmask[node] = 1;
  idxmap[node] = i;
  out_nt[i] = (float)ntype[node];
  out_batch[i] = 0.0f;
}

// ---------- 10: gather x_p = x[perm] * top_vals ----------
__global__ void hgt_gather_x(const float* __restrict__ x, const int* __restrict__ perm,
                             const float* __restrict__ vals, float* out){
  int row = blockIdx.x;
  int node = perm[row];
  float v = vals[row];
  out[(long)row * IN_DIM + threadIdx.x] = x[(long)node * IN_DIM + threadIdx.x] * v;
}

// ---------- 11a: per-block kept-edge counts ----------
__global__ void hgt_ecnt(const int* __restrict__ src, const int* __restrict__ dst,
                         const unsigned char* __restrict__ nmask, int* bcount, int E){
  __shared__ int cnt;
  if (threadIdx.x == 0) cnt = 0;
  __syncthreads();
  int e = blockIdx.x * blockDim.x + threadIdx.x;
  bool pred = (e < E) && nmask[src[e]] && nmask[dst[e]];
  unsigned long long b = __ballot(pred);
  if ((threadIdx.x & 31) == 0) atomicAdd(&cnt, __popcll(b));
  __syncthreads();
  if (threadIdx.x == 0) bcount[blockIdx.x] = cnt;
}

// ---------- 11b: serial exclusive scan of block counts ----------
__global__ void hgt_escan(const int* bcount, int* boff, int nblk){
  if (threadIdx.x == 0 && blockIdx.x == 0){
    int s = 0;
    for (int i = 0; i < nblk; ++i){ boff[i] = s; s += bcount[i]; }
  }
}

// ---------- 11c: compact & remap kept edges (order-preserving) ----------
__global__ void hgt_ecompact(const int* __restrict__ src, const int* __restrict__ dst,
                             const int* __restrict__ etype,
                             const unsigned char* __restrict__ nmask,
                             const int* __restrict__ idxmap, const int* __restrict__ boff,
                             float* out_ei0, float* out_ei1, float* out_et, int E, int Ek){
  __shared__ int woff[8];
  int e = blockIdx.x * blockDim.x + threadIdx.x;
  int s = 0, d = 0, r = 0;
  bool pred = false;
  if (e < E){
    s = src[e]; d = dst[e]; r = etype[e];
    pred = nmask[s] && nmask[d];
  }
  int wid = threadIdx.x >> 5, lane = threadIdx.x & 31;
  unsigned long long b = __ballot(pred);
  if (lane == 0) woff[wid] = __popcll(b);
  __syncthreads();
  if (threadIdx.x == 0){
    int acc = 0;
    for (int w = 0; w < 8; ++w){ int t = woff[w]; woff[w] = acc; acc += t; }
  }
  __syncthreads();
  if (pred){
    int pos = boff[blockIdx.x] + woff[wid] +
              __popcll(b & ((1ull << lane) - 1ull));
    if (pos < Ek){
      out_ei0[pos] = (float)idxmap[s];
      out_ei1[pos] = (float)idxmap[d];
      out_et[pos]  = (float)r;
    }
  }
}

// =====================================================================
extern "C" void kernel_launch(void* const* d_in, const int* in_sizes, int n_in,
                              void* d_out, int out_size, void* d_ws, size_t ws_size,
                              hipStream_t stream){
  const float* x       = (const float*)d_in[0];
  const int*   ei      = (const int*)  d_in[1];
  const int*   ntype   = (const int*)  d_in[2];
  const int*   etype   = (const int*)  d_in[3];
  const float* kW = (const float*)d_in[5];  const float* kb = (const float*)d_in[6];
  const float* qW = (const float*)d_in[7];  const float* qb = (const float*)d_in[8];
  const float* vW = (const float*)d_in[9];  const float* vb = (const float*)d_in[10];
  const float* aW = (const float*)d_in[11]; const float* ab = (const float*)d_in[12];
  const float* mW = (const float*)d_in[13]; const float* mb = (const float*)d_in[14];
  const float* rel_pri = (const float*)d_in[15];
  const float* rel_att = (const float*)d_in[16];
  const float* rel_msg = (const float*)d_in[17];
  const float* skip    = (const float*)d_in[18];

  const int N = in_sizes[0] / IN_DIM;
  const int E = in_sizes[1] / 2;
  const int kkeep = (N + 1) / 2;                       // ceil(0.5*N)
  int Ek = (out_size - kkeep * (IN_DIM + 2)) / 3;      // from concatenated output size
  if (Ek < 0) Ek = 0;
  int SN = 1; while (SN < N) SN <<= 1;                 // sort pad size

  const int* e_src = ei;
  const int* e_dst = ei + E;

  // ---- workspace carve-out ----
  char* p = (char*)d_ws;
  auto alloc = [&](size_t bytes) -> void* {
    void* r = (void*)p; p += (bytes + 255) & ~(size_t)255; return r;
  };
  float* Wall   = (float*)alloc(IN_DIM * 16 * sizeof(float));
  float* bias   = (float*)alloc(16 * sizeof(float));
  float* Kn     = (float*)alloc((size_t)N * sizeof(float));
  float* Qn     = (float*)alloc((size_t)N * sizeof(float));
  float* Vn     = (float*)alloc((size_t)N * sizeof(float));
  float* Mn     = (float*)alloc((size_t)N * sizeof(float));
  unsigned* amaxKey = (unsigned*)alloc((size_t)N * sizeof(unsigned));
  float* den    = (float*)alloc((size_t)N * sizeof(float));
  float* aggr   = (float*)alloc((size_t)N * sizeof(float));
  float* score  = (float*)alloc((size_t)N * sizeof(float));
  float* att    = (float*)alloc((size_t)E * sizeof(float));
  float* msg    = (float*)alloc((size_t)E * sizeof(float));
  unsigned long long* keys = (unsigned long long*)alloc((size_t)SN * sizeof(unsigned long long));
  int*   perm   = (int*)alloc((size_t)kkeep * sizeof(int));
  float* vals   = (float*)alloc((size_t)kkeep * sizeof(float));
  unsigned char* nmask = (unsigned char*)alloc((size_t)N);
  int*   idxmap = (int*)alloc((size_t)N * sizeof(int));
  const int nblk7 = (E + 255) / 256;
  int* bcount = (int*)alloc((size_t)nblk7 * sizeof(int));
  int* boff   = (int*)alloc((size_t)nblk7 * sizeof(int));

  // ---- output layout (floats) ----
  float* out  = (float*)d_out;
  float* o_xp  = out;                                // kkeep*256
  float* o_ei0 = out + (size_t)kkeep * IN_DIM;       // Ek
  float* o_ei1 = o_ei0 + Ek;                         // Ek
  float* o_nt  = o_ei1 + Ek;                         // kkeep
  float* o_et  = o_nt + kkeep;                       // Ek
  float* o_b   = o_et + Ek;                          // kkeep

  // 0/1: setup
  hgt_pack_w<<<(IN_DIM*16 + 255)/256, 256, 0, stream>>>(kW, kb, qW, qb, vW, vb, mW, mb, Wall, bias);
  hgt_init<<<(N + 255)/256, 256, 0, stream>>>(amaxKey, den, aggr, nmask, N);

  // 2: WMMA projection GEMM (N x 256) @ (256 x 16)
  int ntiles = (N + 15) / 16;
  int nblkW  = (ntiles + WPB - 1) / WPB;
  hgt_proj_wmma<<<nblkW, WPB*32, 0, stream>>>(x, ntype, Wall, bias, Kn, Qn, Vn, Mn, N);

  // 3-5: edge passes
  int eblk = (E + 255) / 256;
  hgt_edge1<<<eblk, 256, 0, stream>>>(e_src, e_dst, etype, Kn, Qn, Vn,
                                      rel_att, rel_msg, rel_pri, att, msg, amaxKey, E);
  hgt_edge2<<<eblk, 256, 0, stream>>>(e_dst, att, amaxKey, den, E);
  hgt_edge3<<<eblk, 256, 0, stream>>>(e_dst, att, msg, den, aggr, E);

  // 6: node scores
  hgt_score<<<(N + 255)/256, 256, 0, stream>>>(aggr, ntype, aW, ab, Mn, skip, score, N);

  // 7-8: top-k via full bitonic sort of (score,index) keys, descending
  hgt_fill_keys<<<(SN + 255)/256, 256, 0, stream>>>(score, keys, N, SN);
  for (int k = 2; k <= SN; k <<= 1)
    for (int j = k >> 1; j >= 1; j >>= 1)
      hgt_bitonic<<<SN/256, 256, 0, stream>>>(keys, j, k);

  // 9-10: selection + gather
  hgt_select<<<(kkeep + 255)/256, 256, 0, stream>>>(keys, score, ntype, perm, vals,
                                                    nmask, idxmap, o_nt, o_b, kkeep);
  hgt_gather_x<<<kkeep, IN_DIM, 0, stream>>>(x, perm, vals, o_xp);

  // 11: order-preserving edge compaction + remap
  hgt_ecnt<<<nblk7, 256, 0, stream>>>(e_src, e_dst, nmask, bcount, E);
  hgt_escan<<<1, 1, 0, stream>>>(bcount, boff, nblk7);
  hgt_ecompact<<<nblk7, 256, 0, stream>>>(e_src, e_dst, etype, nmask, idxmap, boff,
                                          o_ei0, o_ei1, o_et, E, Ek);
}